// Temporal_Cosine_Regularizer_48232482734698
// MI455X (gfx1250) — compile-verified
//
#include <hip/hip_runtime.h>
#include <math.h>

// ---------- CDNA5 feature probes (compile-time, reported via #warning) ----------
#if !__has_builtin(__builtin_amdgcn_wmma_f32_16x16x4_f32)
#warning "PROBE MISSING: __builtin_amdgcn_wmma_f32_16x16x4_f32"
#endif
#if __has_builtin(__builtin_amdgcn_global_load_async_to_lds_b128)
#warning "PROBE HAVE: __builtin_amdgcn_global_load_async_to_lds_b128"
#endif
#if __has_builtin(__builtin_amdgcn_cluster_load_async_to_lds_b128)
#warning "PROBE HAVE: __builtin_amdgcn_cluster_load_async_to_lds_b128"
#endif
#if __has_builtin(__builtin_amdgcn_global_load_lds)
#warning "PROBE HAVE: __builtin_amdgcn_global_load_lds"
#endif
#if __has_builtin(__builtin_amdgcn_tensor_load_to_lds)
#warning "PROBE HAVE: __builtin_amdgcn_tensor_load_to_lds"
#endif
#if __has_builtin(__builtin_amdgcn_s_wait_asynccnt)
#warning "PROBE HAVE: __builtin_amdgcn_s_wait_asynccnt"
#endif
#if __has_builtin(__builtin_amdgcn_s_wait_tensorcnt)
#warning "PROBE HAVE: __builtin_amdgcn_s_wait_tensorcnt"
#endif
#if __has_builtin(__builtin_amdgcn_ds_atomic_async_barrier_arrive_b64)
#warning "PROBE HAVE: __builtin_amdgcn_ds_atomic_async_barrier_arrive_b64"
#endif

typedef __attribute__((ext_vector_type(2))) float v2f;
typedef __attribute__((ext_vector_type(8))) float v8f;

#define N_ROWS 65536
#define N_COLS 2048
#define PAIRS (N_ROWS - 1)              // 65535
#define TILE_PAIRS 15                   // 16-row Gram tile -> 15 neighbor pairs
#define NUM_TILES (PAIRS / TILE_PAIRS)  // 4369 (exact)
#define WAVES_PER_BLOCK 8
#define BLOCK_THREADS 256
#define GRID_BLOCKS 547                 // 547*8 = 4376 wave slots >= 4369 tiles
#define TOTAL_WAVES (GRID_BLOCKS * WAVES_PER_BLOCK)
#define NORM_EPS 1e-12f

// One wave per 16-row tile. Gram G = X * X^T via v_wmma_f32_16x16x4_f32 with
// A == B (identical per-lane layout for a Gram operand pair).
// diag(G) = row ||x||^2, superdiag(G) = <x_p, x_{p+1}>.
__global__ __launch_bounds__(BLOCK_THREADS)
void Temporal_Cosine_gram_kernel(const float* __restrict__ factor,
                                 float* __restrict__ partial) {
    __shared__ float scratch[WAVES_PER_BLOCK][16 * 16];

    const int lane = threadIdx.x & 31;
    const int wav  = threadIdx.x >> 5;
    const int gw   = blockIdx.x * WAVES_PER_BLOCK + wav;

    const int m = lane & 15;   // row within tile (A) / col N (B/C)
    const int h = lane >> 4;   // lane-half: selects K sub-block

    float acc = 0.0f;

    for (int tile = gw; tile < NUM_TILES; tile += TOTAL_WAVES) {
        const int row0 = tile * TILE_PAIRS;  // tiles overlap by one row
        const float* rp = factor + (size_t)(row0 + m) * N_COLS + (h << 2);

        v8f c = {};
        // Per 8 K-columns: one b128 load feeds two WMMAs (K-subset order is
        // irrelevant for a Gram accumulation, so .xy / .zw are valid K-quads).
        #pragma unroll 8
        for (int k0 = 0; k0 < N_COLS; k0 += 8) {
            float4 f = *(const float4*)(rp + k0);
            v2f a0 = { f.x, f.y };
            v2f a1 = { f.z, f.w };
            c = __builtin_amdgcn_wmma_f32_16x16x4_f32(
                    false, a0, false, a0, (short)0, c, false, false);
            c = __builtin_amdgcn_wmma_f32_16x16x4_f32(
                    false, a1, false, a1, (short)0, c, false, false);
        }

        // Dump C fragment to LDS per documented layout:
        // VGPR j holds rows M=j (lanes 0-15) and M=j+8 (lanes 16-31), N=lane%16.
        #pragma unroll
        for (int j = 0; j < 8; ++j) {
            scratch[wav][(j + (h << 3)) * 16 + m] = c[j];
        }
        __builtin_amdgcn_wave_barrier();  // keep LDS RW ordering wave-local

        float val = 0.0f;
        if (lane < TILE_PAIRS) {
            const float gpp = scratch[wav][lane * 16 + lane];
            const float gpq = scratch[wav][lane * 16 + lane + 1];
            const float gqq = scratch[wav][(lane + 1) * 16 + (lane + 1)];
            const float np  = fmaxf(sqrtf(gpp), NORM_EPS);
            const float nq  = fmaxf(sqrtf(gqq), NORM_EPS);
            val = gpq / (np * nq);
        }
        // wave32 reduction
        #pragma unroll
        for (int off = 16; off >= 1; off >>= 1)
            val += __shfl_down(val, off, 32);
        acc += val;  // lane 0 holds the tile sum
    }

    if (lane == 0) partial[gw] = acc;  // every slot written each call
}

// Deterministic fixed-order final reduction (single block).
__global__ __launch_bounds__(256)
void Temporal_Cosine_final_reduce(const float* __restrict__ partial,
                                  float* __restrict__ out) {
    __shared__ float s[256];
    float a = 0.0f;
    for (int i = threadIdx.x; i < TOTAL_WAVES; i += 256) a += partial[i];
    s[threadIdx.x] = a;
    __syncthreads();
    #pragma unroll
    for (int off = 128; off >= 1; off >>= 1) {
        if ((int)threadIdx.x < off) s[threadIdx.x] += s[threadIdx.x + off];
        __syncthreads();
    }
    if (threadIdx.x == 0) {
        // -WEIGHT * (positive1 + positive2) = -0.01 * 2 * sum/(N-1)
        out[0] = -0.02f * s[0] / (float)PAIRS;
    }
}

extern "C" void kernel_launch(void* const* d_in, const int* in_sizes, int n_in,
                              void* d_out, int out_size, void* d_ws, size_t ws_size,
                              hipStream_t stream) {
    (void)in_sizes; (void)n_in; (void)out_size; (void)ws_size;
    const float* factor = (const float*)d_in[0];
    float* out = (float*)d_out;
    float* partial = (float*)d_ws;  // TOTAL_WAVES floats (~17.5 KB)

    Temporal_Cosine_gram_kernel<<<GRID_BLOCKS, BLOCK_THREADS, 0, stream>>>(factor, partial);
    Temporal_Cosine_final_reduce<<<1, 256, 0, stream>>>(partial, out);
}